// ColbertModel_34325378629686
// MI455X (gfx1250) — compile-verified
//
#include <hip/hip_runtime.h>
#include <math.h>

typedef __attribute__((ext_vector_type(2))) float v2f;
typedef __attribute__((ext_vector_type(8))) float v8f;

#define L2EPS 1e-12f

__device__ __forceinline__ v8f wmma_f32_4(v2f a, v2f b, v8f c) {
    // D = A(16x4,f32) * B(4x16,f32) + C(16x16,f32)
    return __builtin_amdgcn_wmma_f32_16x16x4_f32(
        /*neg_a=*/false, a, /*neg_b=*/false, b,
        /*c_mod=*/(short)0, c, /*reuse_a=*/false, /*reuse_b=*/false);
}

// ---------------------------------------------------------------------------
// Kernel 0: WT[d][k] = W[k][d]  (768x128 -> 128x768), so the encode kernel's
// B operands become contiguous 8-byte loads.
// ---------------------------------------------------------------------------
__global__ __launch_bounds__(256)
void transpose_w_kernel(const float* __restrict__ W, float* __restrict__ WT)
{
    constexpr int H = 768, D = 128;
    const int idx = blockIdx.x * 256 + threadIdx.x;   // over H*D, coalesced read
    if (idx < H * D) {
        const int k = idx >> 7, d = idx & (D - 1);
        WT[(size_t)d * H + k] = W[idx];
    }
}

// ---------------------------------------------------------------------------
// Kernel 1: col = l2norm((h*mask)[:,1:] @ W + b)
// One wave computes a 16-row x 128-col tile with fp32 WMMA (K-loop over 768).
// A operand: contiguous b64 load + v_pk mask multiply. B operand: contiguous
// b64 load from transposed WT. grid.x = nRows/16 (exact for both encoders).
// ---------------------------------------------------------------------------
__global__ __launch_bounds__(32)
void encode_col_kernel(const float* __restrict__ h, const int* __restrict__ mask,
                       const float* __restrict__ WT, const float* __restrict__ bias,
                       float* __restrict__ colOut, int S, int tokens)
{
    constexpr int H = 768, D = 128, DP = D + 4;   // +4 pad: row stride 132 = 4 mod 64 banks
    __shared__ float tile[16][DP];
    __shared__ float scaleArr[16];

    const int lane   = threadIdx.x;
    const int laneLo = lane & 15;
    const int laneHi = lane >> 4;

    const int rowBase = blockIdx.x * 16;
    const int r     = rowBase + laneLo;          // row owned by this lane (A operand)
    const int batch = r / tokens;
    const int tok   = 1 + (r % tokens);
    const float mv  = (float)mask[batch * S + tok];
    const float* hrow = h + ((size_t)batch * S + tok) * H;

    // Per-lane B column pointers into transposed W.
    const float* wt[8];
#pragma unroll
    for (int t = 0; t < 8; ++t)
        wt[t] = WT + (size_t)(t * 16 + laneLo) * H;

    const v8f z = {};
    v8f acc[8];
#pragma unroll
    for (int t = 0; t < 8; ++t) acc[t] = z;

    for (int k0 = 0; k0 < H; k0 += 4) {
        const int kk = k0 + laneHi * 2;          // lanes 0-15: K={k0,k0+1}; 16-31: {k0+2,k0+3}
        v2f a = *(const v2f*)(hrow + kk);
        a = a * mv;                               // h * mask (v_pk_mul_f32)
#pragma unroll
        for (int t = 0; t < 8; ++t) {
            const v2f bb = *(const v2f*)(wt[t] + kk);
            acc[t] = wmma_f32_4(a, bb, acc[t]);
        }
    }

    // Epilogue: +bias, stage to LDS, per-row L2 normalize, store.
#pragma unroll
    for (int t = 0; t < 8; ++t) {
        const int col = t * 16 + laneLo;
        const float bv = bias[col];
#pragma unroll
        for (int rr = 0; rr < 8; ++rr)
            tile[rr + laneHi * 8][col] = acc[t][rr] + bv;   // VGPR rr -> rows rr / rr+8
    }
    __syncthreads();
    if (lane < 16) {
        float ss = 0.0f;
#pragma unroll 8
        for (int d = 0; d < D; ++d) { float v = tile[lane][d]; ss += v * v; }
        scaleArr[lane] = 1.0f / fmaxf(sqrtf(ss), L2EPS);
    }
    __syncthreads();
    for (int idx = lane; idx < 16 * D; idx += 32) {
        const int m = idx >> 7, d = idx & (D - 1);
        colOut[(size_t)(rowBase + m) * D + d] = tile[m][d] * scaleArr[m];
    }
}

// ---------------------------------------------------------------------------
// Kernel 2: pooled = l2norm(h[:,0] * mask[:,0]); one block per batch element.
// ---------------------------------------------------------------------------
__global__ __launch_bounds__(256)
void pooled_kernel(const float* __restrict__ h, const int* __restrict__ mask,
                   float* __restrict__ out, int S)
{
    constexpr int H = 768;
    __shared__ float red[256];
    const int b = blockIdx.x;
    const int t = threadIdx.x;
    const float mv = (float)mask[(size_t)b * S];
    const float* row = h + (size_t)b * S * H;

    float ss = 0.0f;
    for (int d = t; d < H; d += 256) { float v = row[d] * mv; ss += v * v; }
    red[t] = ss;
    __syncthreads();
    for (int off = 128; off > 0; off >>= 1) {
        if (t < off) red[t] += red[t + off];
        __syncthreads();
    }
    const float scale = 1.0f / fmaxf(sqrtf(red[0]), L2EPS);
    for (int d = t; d < H; d += 256) out[(size_t)b * H + d] = row[d] * mv * scale;
}

// ---------------------------------------------------------------------------
// Kernel 3: sim[q,c] = (1/denom[q]) * sum_i max_j  q_col[q,i] . c_col[c,j]
// Block = (q,c) pair, 4 waves; wave w owns c-token tiles [4w, 4w+4).
// Pad row (i=31) and pad column (j=255) accumulate harmless duplicates and
// are excluded in the epilogue (sum-mask / -inf) instead of per-K masking,
// so every operand is a single contiguous b64 load.
// ---------------------------------------------------------------------------
__global__ __launch_bounds__(128)
void sim_kernel(const float* __restrict__ qcol, const float* __restrict__ ccol,
                const int* __restrict__ qmask, float* __restrict__ sim)
{
    constexpr int D = 128, QT = 31, CT = 255, SQ = 32;
    const int c = blockIdx.x, q = blockIdx.y;
    const int wave = threadIdx.x >> 5;
    const int lane = threadIdx.x & 31;
    const int laneLo = lane & 15, laneHi = lane >> 4;

    const float* qc = qcol + (size_t)q * QT * D;
    const float* cc = ccol + (size_t)c * CT * D;

    const int r0  = laneLo;                        // M-tile 0 row (always valid)
    const int r1c = (16 + laneLo < QT) ? (16 + laneLo) : (QT - 1);  // pad -> dup row 30

    int jIdx[4];
    const float* cj[4];
#pragma unroll
    for (int t = 0; t < 4; ++t) {
        const int j = (wave * 4 + t) * 16 + laneLo;
        jIdx[t] = j;
        cj[t] = cc + (size_t)((j < CT) ? j : (CT - 1)) * D;   // pad -> dup col 254
    }

    const v8f z = {};
    v8f acc[2][4];
#pragma unroll
    for (int mt = 0; mt < 2; ++mt)
#pragma unroll
        for (int t = 0; t < 4; ++t) acc[mt][t] = z;

    for (int k0 = 0; k0 < D; k0 += 4) {
        const int kk = k0 + laneHi * 2;
        const v2f a0 = *(const v2f*)(qc + r0 * D + kk);
        const v2f a1 = *(const v2f*)(qc + r1c * D + kk);
#pragma unroll
        for (int t = 0; t < 4; ++t) {
            const v2f bb = *(const v2f*)(cj[t] + kk);
            acc[0][t] = wmma_f32_4(a0, bb, acc[0][t]);
            acc[1][t] = wmma_f32_4(a1, bb, acc[1][t]);
        }
    }

    // Per-row max over this wave's 64 c-tokens (pad j -> -inf), then cross-wave.
    __shared__ float wmax[4][32];
    const float NEGINF = -3.402823466e38f;
#pragma unroll
    for (int mt = 0; mt < 2; ++mt) {
#pragma unroll
        for (int rr = 0; rr < 8; ++rr) {
            float v = NEGINF;
#pragma unroll
            for (int t = 0; t < 4; ++t) {
                const float x = (jIdx[t] < CT) ? acc[mt][t][rr] : NEGINF;
                v = fmaxf(v, x);
            }
#pragma unroll
            for (int off = 1; off < 16; off <<= 1)
                v = fmaxf(v, __shfl_xor(v, off, 32));    // reduce within each 16-lane half
            if (laneLo == 0)
                wmax[wave][mt * 16 + laneHi * 8 + rr] = v;
        }
    }
    __syncthreads();

    if (wave == 0) {
        const int m = lane;  // row index 0..31
        float g = fmaxf(fmaxf(wmax[0][m], wmax[1][m]), fmaxf(wmax[2][m], wmax[3][m]));
        if (m >= QT) g = 0.0f;                           // pad row excluded from sum
#pragma unroll
        for (int off = 1; off < 32; off <<= 1)
            g += __shfl_xor(g, off, 32);
        if (lane == 0) {
            int denom = 0;
            for (int s = 1; s < SQ; ++s) denom += qmask[q * SQ + s];
            sim[(size_t)q * gridDim.x + c] = g / (float)denom;
        }
    }
}

// ---------------------------------------------------------------------------
extern "C" void kernel_launch(void* const* d_in, const int* in_sizes, int n_in,
                              void* d_out, int out_size, void* d_ws, size_t ws_size,
                              hipStream_t stream)
{
    const float* q_hidden = (const float*)d_in[0];
    const float* c_hidden = (const float*)d_in[1];
    const float* W        = (const float*)d_in[2];
    const float* bias     = (const float*)d_in[3];
    const int*   q_mask   = (const int*)d_in[4];
    const int*   c_mask   = (const int*)d_in[5];
    (void)c_mask; (void)in_sizes; (void)n_in; (void)out_size;

    constexpr int Q = 64, Sq = 32, C = 256, Sc = 256, H = 768, D = 128;
    constexpr int QT = Sq - 1, CT = Sc - 1;

    float* sim      = (float*)d_out;                 // Q*C
    float* q_pooled = sim + (size_t)Q * C;           // Q*H
    float* c_pooled = q_pooled + (size_t)Q * H;      // C*H

    float* WT   = (float*)d_ws;                      // D*H   (transposed W)
    float* qcol = WT + (size_t)D * H;                // Q*QT*D
    float* ccol = qcol + (size_t)Q * QT * D;         // C*CT*D

    transpose_w_kernel<<<(H * D + 255) / 256, 256, 0, stream>>>(W, WT);
    encode_col_kernel<<<(Q * QT) / 16, 32, 0, stream>>>(q_hidden, q_mask, WT, bias, qcol, Sq, QT);
    encode_col_kernel<<<(C * CT) / 16, 32, 0, stream>>>(c_hidden, c_mask, WT, bias, ccol, Sc, CT);
    pooled_kernel<<<Q, 256, 0, stream>>>(q_hidden, q_mask, q_pooled, Sq);
    pooled_kernel<<<C, 256, 0, stream>>>(c_hidden, c_mask, c_pooled, Sc);

    dim3 grid(C, Q);
    sim_kernel<<<grid, 128, 0, stream>>>(qcol, ccol, q_mask, sim);
}